// DClayer_57732950392947
// MI455X (gfx1250) — compile-verified
//
#include <hip/hip_runtime.h>

// Problem constants (match reference).
#define B_  2
#define D_  64
#define H_  160
#define W_  160
#define HP_ 162

static constexpr int TB      = 256;
static constexpr int KS      = D_ * HP_ * W_;      // k-stride in disp  (1,658,880)
static constexpr int BS      = 3 * KS;             // b-stride in disp
static constexpr int N_      = B_ * BS;            // disp elements     (9,953,280)
static constexpr int NV      = B_ * D_ * HP_ * W_; // voxel threads     (3,317,760)
static constexpr int NB_EVAL = NV / TB;            // 12,960 (exact)
static constexpr int NB_N    = N_ / TB;            // 38,880 (exact)
static constexpr int SB      = D_ * H_ * W_;       // src/tar b-stride  (1,638,400)

typedef __attribute__((ext_vector_type(2))) float v2f;
typedef __attribute__((ext_vector_type(8))) float v8f;

// Scalar slots in workspace (first 256 B): Sf[0]=step, Sf[1]=cost; Si[8]=done, Si[9]=accept.

__global__ void init_scalars(float* __restrict__ Sf, int* __restrict__ Si) {
  Sf[0] = 0.1f;   // step
  Sf[1] = 0.0f;   // cost (set by first reduce_update)
  Si[8] = 0;      // done
  Si[9] = 0;      // accept
}

__global__ __launch_bounds__(TB) void seed_kernel(float* __restrict__ disp,
                                                  float* __restrict__ dir,
                                                  const float* __restrict__ dispini) {
  int i = blockIdx.x * TB + threadIdx.x;
  disp[i] = dispini[i];
  dir[i]  = 0.0f;   // => first eval candidate == dispini; first commit: disp unchanged, dir = -g0
}

// Fused cost+grad evaluation at candidate = disp + step*dir (computed on the fly).
// One thread per (b,z,yh,x) over the HP domain; handles all 3 displacement channels.
__global__ __launch_bounds__(TB) void eval_kernel(
    const float* __restrict__ disp, const float* __restrict__ dir,
    const float* __restrict__ tar,  const float* __restrict__ src,
    const float* __restrict__ nout, const float* __restrict__ wp,
    const float* __restrict__ Sf,
    float* __restrict__ gnew, float* __restrict__ partials) {
  int i  = blockIdx.x * TB + threadIdx.x;
  int x  = i % W_;  int t = i / W_;
  int yh = t % HP_; t /= HP_;
  int z  = t % D_;  int b = t / D_;

  const float step = Sf[0];
  const float w    = wp[0];
  const int base   = b * BS + (z * HP_ + yh) * W_ + x;

  // Candidate displacement at this voxel (all 3 channels).
  float dv0 = disp[base         ] + step * dir[base         ];
  float dv1 = disp[base +     KS] + step * dir[base +     KS];
  float dv2 = disp[base + 2 * KS] + step * dir[base + 2 * KS];

  // Regularizer: w * |dv - netoutput|^2 ; grad = 2*w*(dv - netoutput)
  float e0 = dv0 - nout[base];
  float e1 = dv1 - nout[base + KS];
  float e2 = dv2 - nout[base + 2 * KS];
  float g0 = 2.0f * w * e0;
  float g1 = 2.0f * w * e1;
  float g2 = 2.0f * w * e2;
  float lc = w * (e0 * e0 + e1 * e1 + e2 * e2);

  // Data term only for interior H rows (disp[:, :, :, 1:-1, :]).
  if (yh >= 1 && yh <= HP_ - 2) {
    int y = yh - 1;
    float cz = (float)z + dv0;
    float cy = (float)y + dv1;
    float cx = (float)x + dv2;
    // clip gradient mask (1 inside [0, L-1], 0 where clamped)
    float mz = (cz >= 0.0f && cz <= (float)(D_ - 1)) ? 1.0f : 0.0f;
    float my = (cy >= 0.0f && cy <= (float)(H_ - 1)) ? 1.0f : 0.0f;
    float mx = (cx >= 0.0f && cx <= (float)(W_ - 1)) ? 1.0f : 0.0f;
    float zc = fminf(fmaxf(cz, 0.0f), (float)(D_ - 1));
    float yc = fminf(fmaxf(cy, 0.0f), (float)(H_ - 1));
    float xc = fminf(fmaxf(cx, 0.0f), (float)(W_ - 1));
    float z0f = floorf(zc), y0f = floorf(yc), x0f = floorf(xc);
    float wz = zc - z0f, wy = yc - y0f, wx = xc - x0f;
    int z0 = (int)z0f, y0 = (int)y0f, x0 = (int)x0f;
    int z1 = min(z0 + 1, D_ - 1);
    int y1 = min(y0 + 1, H_ - 1);
    int x1 = min(x0 + 1, W_ - 1);

    const float* sp = src + b * SB;
    int zo0 = z0 * (H_ * W_), zo1 = z1 * (H_ * W_);
    int yo0 = y0 * W_,        yo1 = y1 * W_;
    float c000 = sp[zo0 + yo0 + x0], c001 = sp[zo0 + yo0 + x1];
    float c010 = sp[zo0 + yo1 + x0], c011 = sp[zo0 + yo1 + x1];
    float c100 = sp[zo1 + yo0 + x0], c101 = sp[zo1 + yo0 + x1];
    float c110 = sp[zo1 + yo1 + x0], c111 = sp[zo1 + yo1 + x1];

    float dx00 = c001 - c000, dx01 = c011 - c010;
    float dx10 = c101 - c100, dx11 = c111 - c110;
    float c00 = c000 + wx * dx00, c01 = c010 + wx * dx01;
    float c10 = c100 + wx * dx10, c11 = c110 + wx * dx11;
    float c0 = c00 + wy * (c01 - c00);
    float c1 = c10 + wy * (c11 - c10);
    float warped = c0 + wz * (c1 - c0);

    float r = tar[b * SB + (z * H_ + y) * W_ + x] - warped;
    lc += r * r;
    float coef = -2.0f * r;
    float dwz = c1 - c0;
    float dwy = (c01 - c00) + wz * ((c11 - c10) - (c01 - c00));
    float d0x = dx00 + wy * (dx01 - dx00);
    float d1x = dx10 + wy * (dx11 - dx10);
    float dwx = d0x + wz * (d1x - d0x);
    g0 += coef * dwz * mz;
    g1 += coef * dwy * my;
    g2 += coef * dwx * mx;
  }

  gnew[base         ] = g0;
  gnew[base +     KS] = g1;
  gnew[base + 2 * KS] = g2;

  // ---- Wave32 cost reduction via V_WMMA_F32_16X16X4_F32 ones-trick ----
  // A (16x4 f32, 2 VGPR/lane): a.x=lc, a.y=0 => A[m,0]=lc(lane m), A[m,2]=lc(lane m+16).
  // B = all ones (4x16)        => D[m,n] = lc(m) + lc(m+16) for every n.
  // Summing the 8 D VGPRs per lane + one xor-16 shuffle yields the full wave sum.
  v2f a;  a.x = lc;   a.y = 0.0f;
  v2f bo; bo.x = 1.0f; bo.y = 1.0f;
  v8f acc8 = {};
  v8f dd = __builtin_amdgcn_wmma_f32_16x16x4_f32(false, a, false, bo,
                                                 (short)0, acc8, false, false);
  float tsum = dd[0] + dd[1] + dd[2] + dd[3] + dd[4] + dd[5] + dd[6] + dd[7];
  tsum += __shfl_xor(tsum, 16, 32);

  __shared__ float sred[TB / 32];
  int lane = threadIdx.x & 31;
  int wid  = threadIdx.x >> 5;
  if (lane == 0) sred[wid] = tsum;
  __syncthreads();
  if (threadIdx.x == 0) {
    float s = 0.0f;
    #pragma unroll
    for (int k = 0; k < TB / 32; ++k) s += sred[k];
    partials[blockIdx.x] = s;   // deterministic: no atomics
  }
}

// Deterministic final reduction fused with the line-search state update.
__global__ __launch_bounds__(256) void reduce_update(const float* __restrict__ partials,
                                                     float* __restrict__ Sf,
                                                     int* __restrict__ Si, int first) {
  __shared__ float sm[256];
  float s = 0.0f;
  for (int j = threadIdx.x; j < NB_EVAL; j += 256) s += partials[j];
  sm[threadIdx.x] = s;
  __syncthreads();
  for (int off = 128; off > 0; off >>= 1) {
    if (threadIdx.x < off) sm[threadIdx.x] += sm[threadIdx.x + off];
    __syncthreads();
  }
  if (threadIdx.x == 0) {
    float cn = sm[0];
    if (first) {                    // initial eval: unconditional accept of -g0, cost = c0
      Sf[1] = cn;
      Si[9] = 1;
    } else {
      float c  = Sf[1];
      float st = Sf[0];
      int   dn = Si[8];
      bool worse  = cn > c;
      bool halve  = (!dn) && worse && (st > 1.0e-4f);
      bool accept = (!dn) && (!worse);
      int  done2  = dn || (worse && (st <= 1.0e-4f));
      Sf[0] = halve ? st * 0.5f : st;   // step unchanged on accept => commit uses eval's step
      if (accept) Sf[1] = cn;
      Si[8] = done2;
      Si[9] = accept ? 1 : 0;
    }
  }
}

__global__ __launch_bounds__(TB) void commit_kernel(float* __restrict__ disp,
                                                    float* __restrict__ dir,
                                                    const float* __restrict__ gnew,
                                                    const float* __restrict__ Sf,
                                                    const int* __restrict__ Si) {
  if (!Si[9]) return;               // uniform branch
  int i = blockIdx.x * TB + threadIdx.x;
  float st = Sf[0];
  disp[i] = disp[i] + st * dir[i];  // accepted candidate
  dir[i]  = -gnew[i];               // new search direction
}

extern "C" void kernel_launch(void* const* d_in, const int* in_sizes, int n_in,
                              void* d_out, int out_size, void* d_ws, size_t ws_size,
                              hipStream_t stream) {
  const float* dispini = (const float*)d_in[0];
  const float* tar     = (const float*)d_in[1];
  const float* src     = (const float*)d_in[2];
  const float* nout    = (const float*)d_in[3];
  // d_in[4] = grid: identity by construction, recomputed from indices on device.
  const float* wgt     = (const float*)d_in[5];

  float* disp = (float*)d_out;          // evolving disp lives in the output buffer

  // Workspace layout: [64 scalar floats][16384 partials][dir: N_][gnew: N_] ~= 80 MB.
  float* Sf       = (float*)d_ws;
  int*   Si       = (int*)d_ws;
  float* partials = Sf + 64;
  float* dir      = partials + 16384;
  float* gnew     = dir + N_;

  init_scalars<<<1, 1, 0, stream>>>(Sf, Si);
  seed_kernel<<<NB_N, TB, 0, stream>>>(disp, dir, dispini);

  // 1 initial eval + 10 line-search steps, all resident on the GPU (graph-capturable).
  for (int it = 0; it < 11; ++it) {
    eval_kernel<<<NB_EVAL, TB, 0, stream>>>(disp, dir, tar, src, nout, wgt,
                                            Sf, gnew, partials);
    reduce_update<<<1, 256, 0, stream>>>(partials, Sf, Si, it == 0 ? 1 : 0);
    commit_kernel<<<NB_N, TB, 0, stream>>>(disp, dir, gnew, Sf, Si);
  }
}